// MaxLocaldist_50551765074334
// MI455X (gfx1250) — compile-verified
//
#include <hip/hip_runtime.h>
#include <stdint.h>

// MaxLocaldist on MI455X (gfx1250):
//   out(b,0,h,w) = sqrt( max over 36 pairs of |p_i - p_j|^2 ) for the 3x3
//   window of 3-channel pixels around (h,w); border ring = 0.
// VALU-bound; tile staged through LDS by the CDNA5 Tensor Data Mover
// (one tensor_load_to_lds per block, TENSORcnt-tracked). Each thread
// computes TWO vertically adjacent outputs: their windows share 15 of the
// pairwise distances (57 distinct pairs instead of 72) and 18 LDS values.

#define W_IMG 224
#define H_IMG 224
#define C_IMG 3
#define BX 32
#define BYO 8            // output rows per block
#define BY 4             // thread rows (2 outputs per thread)
#define TILE_X (BX + 2)  // 34
#define TILE_Y (BYO + 2) // 10
#define PLANE (W_IMG * H_IMG)   // 50176

typedef unsigned int u32x4 __attribute__((ext_vector_type(4)));
typedef int          i32x8 __attribute__((ext_vector_type(8)));
typedef int          i32x4 __attribute__((ext_vector_type(4)));

__global__ __launch_bounds__(BX * BY)
void maxlocaldist_kernel(const float* __restrict__ x, float* __restrict__ out)
{
    // Contiguous LDS tile, layout [c][y][x] (x fastest) == TDM write order.
    __shared__ float tile[C_IMG][TILE_Y][TILE_X];   // 1020 floats = 4080 B

    const int bi = blockIdx.z;
    const int w0 = blockIdx.x * BX;
    const int h0 = blockIdx.y * BYO;

    // Clamp tile origin so the whole (TILE_Y x TILE_X) footprint is in-bounds.
    int ox = w0 - 1; if (ox < 0) ox = 0; if (ox > W_IMG - TILE_X) ox = W_IMG - TILE_X;
    int oy = h0 - 1; if (oy < 0) oy = 0; if (oy > H_IMG - TILE_Y) oy = H_IMG - TILE_Y;

    const size_t base = (size_t)bi * (C_IMG * PLANE);

#if defined(__gfx1250__) && __has_builtin(__builtin_amdgcn_tensor_load_to_lds)
    // ---- CDNA5 Tensor Data Mover: DMA the 3x10x34 fp32 tile into LDS. ----
    if (threadIdx.y == 0) {   // wave 0 issues the TDM op
        const uint64_t ga  = (uint64_t)(uintptr_t)(x + base + (size_t)oy * W_IMG + ox);
        const uint32_t lds = (uint32_t)(uintptr_t)&tile[0][0][0];

        // D# group 0 (128b): count=1 | lds_addr | global_addr[56:0] | type=2
        u32x4 g0;
        g0[0] = 1u;
        g0[1] = lds;
        g0[2] = (uint32_t)ga;
        g0[3] = (uint32_t)((ga >> 32) & 0x1FFFFFFu) | 0x80000000u; // type=2 in [127:126]

        // D# group 1 (256b): data_size=4B, tensor dims 224x224(x3 via dim2),
        // tile 34x10x3, dim0_stride=224 (row), dim1_stride=50176 (channel plane).
        i32x8 g1;
        g1[0] = (int)(2u << 16);                          // data_size=2 (4 bytes)
        g1[1] = (int)((uint32_t)W_IMG << 16);             // tensor_dim0[15:0] in [31:16]
        g1[2] = (int)((uint32_t)H_IMG << 16);             // dim0 hi=0 | tensor_dim1[15:0]
        g1[3] = (int)((uint32_t)TILE_X << 16);            // dim1 hi=0 | tile_dim0
        g1[4] = (int)((uint32_t)TILE_Y | ((uint32_t)C_IMG << 16)); // tile_dim1|tile_dim2
        g1[5] = (int)W_IMG;                               // tensor_dim0_stride[31:0]
        g1[6] = (int)(((uint32_t)PLANE & 0xFFFFu) << 16); // stride0 hi=0 | stride1[15:0]
        g1[7] = (int)((uint32_t)PLANE >> 16);             // stride1[47:16]

        // D# group 2: tensor_dim2=3 (channels); no dim3.
        i32x4 g2; g2[0] = C_IMG; g2[1] = 0; g2[2] = 0; g2[3] = 0;
        i32x4 g3; g3[0] = 0; g3[1] = 0; g3[2] = 0; g3[3] = 0;
        // VADDR4 slot is unused per ISA -> zeros.
        i32x8 g4; g4[0] = 0; g4[1] = 0; g4[2] = 0; g4[3] = 0;
                  g4[4] = 0; g4[5] = 0; g4[6] = 0; g4[7] = 0;

        __builtin_amdgcn_tensor_load_to_lds(g0, g1, g2, g3, g4, 0);
        __builtin_amdgcn_s_wait_tensorcnt(0);
    }
    __syncthreads();
#else
    // ---- Fallback: cooperative loads into LDS. ----
    {
        const int tid = threadIdx.y * BX + threadIdx.x;
        float* lp = &tile[0][0][0];
        const int total = C_IMG * TILE_Y * TILE_X;   // 1020
        for (int s = tid; s < total; s += BX * BY) {
            const int c  = s / (TILE_Y * TILE_X);
            const int r  = (s / TILE_X) % TILE_Y;
            const int cx = s % TILE_X;
            lp[s] = x[base + (size_t)c * PLANE + (size_t)(oy + r) * W_IMG + (ox + cx)];
        }
    }
    __syncthreads();
#endif

    const int w  = w0 + (int)threadIdx.x;
    const int hA = h0 + 2 * (int)threadIdx.y;   // first output row
    const int hB = hA + 1;                      // second output row

    const bool wOK = (w >= 1) && (w <= W_IMG - 2);
    const bool iA  = wOK && (hA >= 1) && (hA <= H_IMG - 2);
    const bool iB  = wOK && (hB <= H_IMG - 2);  // hB >= 1 always

    float resA = 0.0f, resB = 0.0f;
    if (iA || iB) {
        const int lx = (w - 1) - ox;   // left column of windows in tile

        // Load 4 input rows (hA-1 .. hA+2) x 3 cols x 3 channels.
        // Rows outside the tile (only at image borders) are clamped; the
        // windows that would use them are discarded via iA/iB.
        float v0[12], v1[12], v2[12];   // per channel, patch p = 3*r + j
#pragma unroll
        for (int r = 0; r < 4; ++r) {
            int lyr = (hA - 1 + r) - oy;
            if (lyr < 0) lyr = 0;
            if (lyr > TILE_Y - 1) lyr = TILE_Y - 1;
#pragma unroll
            for (int j = 0; j < 3; ++j) {
                v0[3 * r + j] = tile[0][lyr][lx + j];
                v1[3 * r + j] = tile[1][lyr][lx + j];
                v2[3 * r + j] = tile[2][lyr][lx + j];
            }
        }

        // Window A = patches 0..8 (rows 0..2); window B = patches 3..11.
        // 57 distinct pairs; 15 shared. sqrt is monotone -> one sqrt per
        // output instead of 36.
        float mA = 0.0f, mB = 0.0f;
#pragma unroll
        for (int a = 0; a < 12; ++a) {
#pragma unroll
            for (int q = a + 1; q < 12; ++q) {
                const bool inA = (q < 9);        // a < 9 implied
                const bool inB = (a >= 3);       // q >= 3 implied
                if (!inA && !inB) continue;      // (row0, row3) pairs: skip
                const float d0 = v0[a] - v0[q];
                const float d1 = v1[a] - v1[q];
                const float d2 = v2[a] - v2[q];
                const float s  = __builtin_fmaf(d0, d0, __builtin_fmaf(d1, d1, d2 * d2));
                if (inA) mA = fmaxf(mA, s);
                if (inB) mB = fmaxf(mB, s);
            }
        }
        resA = iA ? __builtin_sqrtf(mA) : 0.0f;
        resB = iB ? __builtin_sqrtf(mB) : 0.0f;
    }

    float* op = out + (size_t)bi * PLANE + (size_t)hA * W_IMG + w;
    op[0]     = resA;
    op[W_IMG] = resB;
}

extern "C" void kernel_launch(void* const* d_in, const int* in_sizes, int n_in,
                              void* d_out, int out_size, void* d_ws, size_t ws_size,
                              hipStream_t stream)
{
    (void)in_sizes; (void)n_in; (void)out_size; (void)d_ws; (void)ws_size;
    const float* x = (const float*)d_in[0];
    float* out = (float*)d_out;

    dim3 block(BX, BY, 1);
    dim3 grid(W_IMG / BX, H_IMG / BYO, 32);   // 7 x 28 x 32
    maxlocaldist_kernel<<<grid, block, 0, stream>>>(x, out);
}